// Sampler_35502199669440
// MI455X (gfx1250) — compile-verified
//
#include <hip/hip_runtime.h>
#include <stdint.h>

typedef float v4f __attribute__((ext_vector_type(4)));

#define TPB        1024
#define NCOLS      128000
#define NBINS      4096
#define PIPE       4
#define TILE_ELEMS (TPB * 4)                 // 4096 floats = 16 KB / tile
#define FULL_TILES (NCOLS / TILE_ELEMS)      // 31
#define TAIL_BASE  (FULL_TILES * TILE_ELEMS) // 126976, tail = 1024 == TPB
#define CAND_CAP   512
#define TEMP       0.8f
#define P_TOP      0.9f

// Order-preserving float <-> uint32 key (standard radix trick).
__device__ __forceinline__ uint32_t f2key(float f) {
  uint32_t u = __float_as_uint(f);
  return (u & 0x80000000u) ? ~u : (u | 0x80000000u);
}
__device__ __forceinline__ float key2f(uint32_t k) {
  uint32_t u = (k & 0x80000000u) ? (k & 0x7fffffffu) : ~k;
  return __uint_as_float(u);
}

__global__ __launch_bounds__(TPB, 1) void topk_topp_kernel(
    const float* __restrict__ logits, const int* __restrict__ top_k,
    float* __restrict__ out) {
  // LDS: 64 KB tiles + 32 KB histogram + 4 KB candidates (~100 KB / 320 KB,
  // so up to 3 workgroups can co-reside on a WGP).
  __shared__ __align__(16) float tileBuf[PIPE][TILE_ELEMS];
  __shared__ uint32_t hist[NBINS][2];   // 2 interleaved copies (adjacent banks)
  __shared__ float candV[CAND_CAP];
  __shared__ float candE[CAND_CAP];
  __shared__ uint32_t s_count, s_maxKey, s_binB, s_cutKey;
  __shared__ float s_Z;

  const int tid = threadIdx.x;
  const int row = blockIdx.x;
  const float* rowIn  = logits + (size_t)row * NCOLS;
  float*       rowOut = out    + (size_t)row * NCOLS;
  const int copy = tid & 1;

  for (int i = tid; i < NBINS; i += TPB) { hist[i][0] = 0u; hist[i][1] = 0u; }
  if (tid == 0) {
    s_count = 0u; s_maxKey = 0u; s_binB = 0u; s_cutKey = 0xFFFFFFFFu; s_Z = 0.0f;
  }
  __syncthreads();

  // ---------- Pass 1: async-stream row -> LDS, 4096-bin histogram ----------
  // Depth-4 global_load_async_to_lds pipeline. Each thread consumes exactly
  // the float4 it requested, so a per-wave s_wait_asynccnt is sufficient —
  // no workgroup barriers inside the streaming loop. Row max key is tracked
  // in a register for free (becomes both bin-search start and softmax m).
  auto issue = [&](int t) {
    if (t < FULL_TILES) {
      uint32_t ldsOff = (uint32_t)(uintptr_t)&tileBuf[t % PIPE][tid * 4];
      uint32_t gOff   = (uint32_t)((t * TILE_ELEMS + tid * 4) * 4);
      asm volatile("global_load_async_to_lds_b128 %0, %1, %2"
                   :: "v"(ldsOff), "v"(gOff), "s"(rowIn)
                   : "memory");
    }
  };

  for (int t = 0; t < PIPE; ++t) issue(t);

  uint32_t kmax = 0u;
  for (int t = 0; t < FULL_TILES; ++t) {
    if (t < FULL_TILES - (PIPE - 1))
      asm volatile("s_wait_asynccnt 3" ::: "memory");
    else
      asm volatile("s_wait_asynccnt 0" ::: "memory");
    v4f q = *reinterpret_cast<const v4f*>(&tileBuf[t % PIPE][tid * 4]);
#pragma unroll
    for (int c = 0; c < 4; ++c) {
      float y = q[c] / TEMP;
      uint32_t key = f2key(y);
      if (key > kmax) kmax = key;
      atomicAdd(&hist[key >> 20][copy], 1u);
    }
    issue(t + PIPE);
  }
  { // tail: exactly TPB leftover elements, one per thread
    float y = rowIn[TAIL_BASE + tid] / TEMP;
    uint32_t key = f2key(y);
    if (key > kmax) kmax = key;
    atomicAdd(&hist[key >> 20][copy], 1u);
  }
  atomicMax(&s_maxKey, kmax);   // ds_max_u32, once per thread
  __syncthreads();

  // ---------- Find bin B holding the K-th largest -------------------------
  // Walk down from the max bin; top bins are sparse so this is ~tens of
  // serial iterations (worst-case NBINS), replacing a 12-barrier LDS scan.
  const int K = min(max(top_k[0], 1), CAND_CAP - 1);
  if (tid == 0) {
    int b = (int)(s_maxKey >> 20);
    uint32_t acc = 0u;
    for (;;) {
      acc += hist[b][0] + hist[b][1];
      if (acc >= (uint32_t)K || b == 0) break;
      --b;
    }
    s_binB = (uint32_t)b;
  }
  __syncthreads();
  const uint32_t binB = s_binB;

  // ---------- Pass 2 (L2-resident re-read): collect candidates >= bin B ----
  const v4f* row4 = reinterpret_cast<const v4f*>(rowIn);
  for (int i = tid; i < NCOLS / 4; i += TPB) {
    v4f q = row4[i];
#pragma unroll
    for (int c = 0; c < 4; ++c) {
      float y = q[c] / TEMP;
      uint32_t key = f2key(y);
      if ((key >> 20) >= binB) {
        uint32_t pos = atomicAdd(&s_count, 1u);
        if (pos < CAND_CAP) candV[pos] = y;
      }
    }
  }
  __syncthreads();
  const int   C = (int)min(s_count, (uint32_t)CAND_CAP);
  const float m = key2f(s_maxKey);
  if (tid < C) candE[tid] = expf(candV[tid] - m);
  __syncthreads();

  // ---------- Rank + exclusive exp-prefix (no sort needed) -----------------
  // Kept set is a prefix of the descending sort: token v survives top-p iff
  // sum_{u>v} exp(u-m) <= p * Z (or v is the max). cutoff = min kept value.
  int r_i = 0x7fffffff; float g_i = 0.0f;
  if (tid < C) {
    const float v = candV[tid];
    int r = 0; float g = 0.0f;
    for (int j = 0; j < C; ++j) {
      float vj = candV[j];
      if (vj > v) { ++r; g += candE[j]; }
    }
    r_i = r; g_i = g;
    if (r < K) atomicAdd(&s_Z, candE[tid]);   // ds_add_f32
  }
  __syncthreads();
  if (r_i < K) {
    if (r_i == 0 || g_i <= P_TOP * s_Z)
      atomicMin(&s_cutKey, f2key(candV[tid]));  // ds_min_u32
  }
  __syncthreads();
  const float cut = key2f(s_cutKey);

  // ---------- Pass 3: rewrite row (non-temporal: protect L2 residency) -----
  v4f* out4 = reinterpret_cast<v4f*>(rowOut);
  const float NEGINF = __int_as_float(0xff800000);
  for (int i = tid; i < NCOLS / 4; i += TPB) {
    v4f q = __builtin_nontemporal_load(&row4[i]);
    v4f o;
#pragma unroll
    for (int c = 0; c < 4; ++c) {
      float y = q[c] / TEMP;
      o[c] = (y >= cut) ? y : NEGINF;
    }
    __builtin_nontemporal_store(o, &out4[i]);
  }
}

extern "C" void kernel_launch(void* const* d_in, const int* in_sizes, int n_in,
                              void* d_out, int out_size, void* d_ws, size_t ws_size,
                              hipStream_t stream) {
  const float* logits = (const float*)d_in[0];
  const int*   top_k  = (const int*)d_in[1];
  float*       out    = (float*)d_out;
  const int rows = in_sizes[0] / NCOLS;   // 256
  topk_topp_kernel<<<rows, TPB, 0, stream>>>(logits, top_k, out);
}